// AttentionMemoryInterface_26276609917646
// MI455X (gfx1250) — compile-verified
//
#include <hip/hip_runtime.h>
#include <hip/hip_bf16.h>

#define BATCH   64
#define IN_DIM  256
#define MEM_DIM 128
#define N_SLOTS 4096

typedef float v2f __attribute__((ext_vector_type(2)));
typedef float v8f __attribute__((ext_vector_type(8)));

// ---------------------------------------------------------------------------
// Kernel 1: fused projections  x[64,256] @ [W_write | W_wq | W_rq][256,128]x3
// One wave (32 threads) per 16x16 output tile, K=256 in steps of 4 via
// v_wmma_f32_16x16x4_f32.  Grid: (64/16) * (384/16) = 96 waves.
// ---------------------------------------------------------------------------
__global__ __launch_bounds__(32)
void proj_wmma_kernel(const float* __restrict__ x,
                      const float* __restrict__ Ww, const float* __restrict__ bw,
                      const float* __restrict__ Wq, const float* __restrict__ bq,
                      const float* __restrict__ Wr, const float* __restrict__ br,
                      float* __restrict__ mv, float* __restrict__ wq,
                      float* __restrict__ rq)
{
    const int lane = threadIdx.x;          // 0..31
    const int tile = blockIdx.x;           // 0..95
    const int mt = tile / 24;              // 4 M tiles
    const int nt = tile % 24;              // 24 N tiles over 384 cols
    const int m0 = mt * 16;
    const int n0 = nt * 16;                // global column in [0,384)
    const int wsel = n0 >> 7;              // which weight matrix
    const int col0 = n0 & 127;             // column inside that matrix

    const float* W;  const float* bias;  float* outp;
    if (wsel == 0)      { W = Ww; bias = bw; outp = mv; }
    else if (wsel == 1) { W = Wq; bias = bq; outp = wq; }
    else                { W = Wr; bias = br; outp = rq; }

    const int half = lane >> 4;            // 0: K lo pair, 1: K hi pair
    const int l16  = lane & 15;
    const int row  = m0 + l16;             // A: lane = M row
    const int col  = col0 + l16;           // B: lane = N col

    v8f c = {};                            // 16x16 f32 accumulator (8 VGPRs)
    for (int k = 0; k < IN_DIM; k += 4) {
        v2f a, b;
        // A 16x4: lanes 0-15 hold K={k,k+1}, lanes 16-31 hold K={k+2,k+3}
        a.x = x[row * IN_DIM + k + 2 * half];
        a.y = x[row * IN_DIM + k + 2 * half + 1];
        // B 4x16 (K x N): mirrored layout, lane = N
        b.x = W[(k + 2 * half)     * MEM_DIM + col];
        b.y = W[(k + 2 * half + 1) * MEM_DIM + col];
        c = __builtin_amdgcn_wmma_f32_16x16x4_f32(
                /*neg_a=*/false, a, /*neg_b=*/false, b,
                /*c_mod=*/(short)0, c, /*reuse_a=*/false, /*reuse_b=*/false);
    }
    // D layout: VGPR j -> row m0+j (lanes 0-15) / m0+j+8 (lanes 16-31), N = l16
    const float bcol = bias[col0 + l16];
#pragma unroll
    for (int j = 0; j < 8; ++j) {
        const int mrow = m0 + j + 8 * half;
        outp[mrow * MEM_DIM + col0 + l16] = c[j] + bcol;
    }
}

// ---------------------------------------------------------------------------
// Kernel 2: write_logits[b,n] = dot(memory[b,n,:], q[b,:])
// Full wave per slot: 32 lanes x float4 = 128 dims, shuffle reduce.
// Grid: 64 batches * 16 chunks, 256 threads (8 waves, 32 slots each).
// ---------------------------------------------------------------------------
__global__ __launch_bounds__(256)
void wlogits_kernel(const float* __restrict__ mem, const float* __restrict__ q,
                    float* __restrict__ logits)
{
    const int b     = blockIdx.x >> 4;
    const int chunk = blockIdx.x & 15;
    const int wave  = threadIdx.x >> 5;
    const int lane  = threadIdx.x & 31;

    const float4 q4 = reinterpret_cast<const float4*>(q + (size_t)b * MEM_DIM)[lane];
    const size_t base = (size_t)b * N_SLOTS * MEM_DIM;
    const int slot0 = chunk * 256 + wave * 32;

    for (int s = 0; s < 32; ++s) {
        const int n = slot0 + s;
        const float4 m4 =
            reinterpret_cast<const float4*>(mem + base + (size_t)n * MEM_DIM)[lane];
        float p = m4.x * q4.x + m4.y * q4.y + m4.z * q4.z + m4.w * q4.w;
#pragma unroll
        for (int off = 16; off; off >>= 1) p += __shfl_xor(p, off, 32);
        if (lane == 0) logits[b * N_SLOTS + n] = p;
    }
}

// ---------------------------------------------------------------------------
// Kernel 3: per-batch softmax stats (max, sum(exp(l-max))). 64 blocks x 256.
// ---------------------------------------------------------------------------
__global__ __launch_bounds__(256)
void softmax_stats_kernel(const float* __restrict__ logits, float* __restrict__ stat)
{
    __shared__ float red[256];
    const int b = blockIdx.x;
    const int t = threadIdx.x;

    float m = -INFINITY;
    for (int n = t; n < N_SLOTS; n += 256) m = fmaxf(m, logits[b * N_SLOTS + n]);
    red[t] = m; __syncthreads();
    for (int s = 128; s; s >>= 1) {
        if (t < s) red[t] = fmaxf(red[t], red[t + s]);
        __syncthreads();
    }
    const float mx = red[0]; __syncthreads();

    float sum = 0.f;
    for (int n = t; n < N_SLOTS; n += 256)
        sum += __expf(logits[b * N_SLOTS + n] - mx);
    red[t] = sum; __syncthreads();
    for (int s = 128; s; s >>= 1) {
        if (t < s) red[t] += red[t + s];
        __syncthreads();
    }
    if (t == 0) { stat[2 * b] = mx; stat[2 * b + 1] = red[0]; }
}

// ---------------------------------------------------------------------------
// Kernel 4: read_logits[b,n] = dot(new_mem[b,n,:], rq[b,:]) with
// new_mem recomputed on the fly (never materialized to HBM).
// ---------------------------------------------------------------------------
__global__ __launch_bounds__(256)
void rlogits_kernel(const float* __restrict__ mem, const float* __restrict__ mv,
                    const float* __restrict__ rq,  const float* __restrict__ wl,
                    const float* __restrict__ wstat, float* __restrict__ rl)
{
    const int b     = blockIdx.x >> 4;
    const int chunk = blockIdx.x & 15;
    const int wave  = threadIdx.x >> 5;
    const int lane  = threadIdx.x & 31;

    const float4 q4  = reinterpret_cast<const float4*>(rq + (size_t)b * MEM_DIM)[lane];
    const float4 mv4 = reinterpret_cast<const float4*>(mv + (size_t)b * MEM_DIM)[lane];
    const float  wmax = wstat[2 * b];
    const float  winv = 1.0f / wstat[2 * b + 1];
    const size_t base = (size_t)b * N_SLOTS * MEM_DIM;
    const int slot0 = chunk * 256 + wave * 32;

    for (int s = 0; s < 32; ++s) {
        const int n = slot0 + s;
        const float w = __expf(wl[b * N_SLOTS + n] - wmax) * winv;
        const float4 m4 =
            reinterpret_cast<const float4*>(mem + base + (size_t)n * MEM_DIM)[lane];
        float4 nm;
        nm.x = m4.x + w * (mv4.x - m4.x);
        nm.y = m4.y + w * (mv4.y - m4.y);
        nm.z = m4.z + w * (mv4.z - m4.z);
        nm.w = m4.w + w * (mv4.w - m4.w);
        float p = nm.x * q4.x + nm.y * q4.y + nm.z * q4.z + nm.w * q4.w;
#pragma unroll
        for (int off = 16; off; off >>= 1) p += __shfl_xor(p, off, 32);
        if (lane == 0) rl[b * N_SLOTS + n] = p;
    }
}

// ---------------------------------------------------------------------------
// Kernel 5: read_out[b,:] = sum_n rw[b,n] * new_mem[b,n,:]
// 64 blocks x 256 threads; 8 slot-groups accumulate in registers, LDS combine.
// ---------------------------------------------------------------------------
__global__ __launch_bounds__(256)
void readout_kernel(const float* __restrict__ mem, const float* __restrict__ mv,
                    const float* __restrict__ wl,  const float* __restrict__ wstat,
                    const float* __restrict__ rl,  const float* __restrict__ rstat,
                    float* __restrict__ ro)
{
    __shared__ float lds[8 * MEM_DIM];
    const int b    = blockIdx.x;
    const int g    = threadIdx.x >> 5;   // slot group 0..7
    const int lane = threadIdx.x & 31;

    const float4 mv4 = reinterpret_cast<const float4*>(mv + (size_t)b * MEM_DIM)[lane];
    const float wmax = wstat[2 * b], winv = 1.0f / wstat[2 * b + 1];
    const float rmax = rstat[2 * b], rinv = 1.0f / rstat[2 * b + 1];
    const size_t base = (size_t)b * N_SLOTS * MEM_DIM;

    float4 acc = make_float4(0.f, 0.f, 0.f, 0.f);
    for (int n = g; n < N_SLOTS; n += 8) {
        const float w = __expf(wl[b * N_SLOTS + n] - wmax) * winv;
        const float r = __expf(rl[b * N_SLOTS + n] - rmax) * rinv;
        const float4 m4 =
            reinterpret_cast<const float4*>(mem + base + (size_t)n * MEM_DIM)[lane];
        acc.x += r * (m4.x + w * (mv4.x - m4.x));
        acc.y += r * (m4.y + w * (mv4.y - m4.y));
        acc.z += r * (m4.z + w * (mv4.z - m4.z));
        acc.w += r * (m4.w + w * (mv4.w - m4.w));
    }
    lds[g * MEM_DIM + 4 * lane + 0] = acc.x;
    lds[g * MEM_DIM + 4 * lane + 1] = acc.y;
    lds[g * MEM_DIM + 4 * lane + 2] = acc.z;
    lds[g * MEM_DIM + 4 * lane + 3] = acc.w;
    __syncthreads();
    if (threadIdx.x < MEM_DIM) {
        float s = 0.f;
#pragma unroll
        for (int gg = 0; gg < 8; ++gg) s += lds[gg * MEM_DIM + threadIdx.x];
        ro[b * MEM_DIM + threadIdx.x] = s;
    }
}

// ---------------------------------------------------------------------------
// Kernel 6: out = read_out[64,128] @ W_ro[128,256] + b_ro  (WMMA f32 16x16x4)
// Grid: (64/16)*(256/16) = 64 waves of 32 threads.
// ---------------------------------------------------------------------------
__global__ __launch_bounds__(32)
void out_wmma_kernel(const float* __restrict__ ro, const float* __restrict__ Wro,
                     const float* __restrict__ bro, float* __restrict__ out)
{
    const int lane = threadIdx.x;
    const int tile = blockIdx.x;         // 0..63
    const int mt = tile / 16;
    const int nt = tile % 16;
    const int m0 = mt * 16;
    const int n0 = nt * 16;

    const int half = lane >> 4;
    const int l16  = lane & 15;
    const int row  = m0 + l16;
    const int col  = n0 + l16;

    v8f c = {};
    for (int k = 0; k < MEM_DIM; k += 4) {
        v2f a, b;
        a.x = ro[row * MEM_DIM + k + 2 * half];
        a.y = ro[row * MEM_DIM + k + 2 * half + 1];
        b.x = Wro[(k + 2 * half)     * IN_DIM + col];
        b.y = Wro[(k + 2 * half + 1) * IN_DIM + col];
        c = __builtin_amdgcn_wmma_f32_16x16x4_f32(
                false, a, false, b, (short)0, c, false, false);
    }
    const float bcol = bro[n0 + l16];
#pragma unroll
    for (int j = 0; j < 8; ++j) {
        const int mrow = m0 + j + 8 * half;
        out[mrow * IN_DIM + n0 + l16] = c[j] + bcol;
    }
}

// ---------------------------------------------------------------------------
extern "C" void kernel_launch(void* const* d_in, const int* in_sizes, int n_in,
                              void* d_out, int out_size, void* d_ws, size_t ws_size,
                              hipStream_t stream)
{
    const float* x       = (const float*)d_in[0];
    const float* memory  = (const float*)d_in[1];
    const float* W_write = (const float*)d_in[2];
    const float* b_write = (const float*)d_in[3];
    const float* W_wq    = (const float*)d_in[4];
    const float* b_wq    = (const float*)d_in[5];
    const float* W_rq    = (const float*)d_in[6];
    const float* b_rq    = (const float*)d_in[7];
    const float* W_ro    = (const float*)d_in[8];
    const float* b_ro    = (const float*)d_in[9];
    float* out = (float*)d_out;

    // workspace layout (floats)
    float* ws    = (float*)d_ws;
    float* mv    = ws;                       // 64*128
    float* wq    = mv + BATCH * MEM_DIM;     // 64*128
    float* rq    = wq + BATCH * MEM_DIM;     // 64*128
    float* wl    = rq + BATCH * MEM_DIM;     // 64*4096
    float* rl    = wl + BATCH * N_SLOTS;     // 64*4096
    float* wstat = rl + BATCH * N_SLOTS;     // 64*2
    float* rstat = wstat + 2 * BATCH;        // 64*2
    float* ro    = rstat + 2 * BATCH;        // 64*128

    // 1) projections via WMMA
    proj_wmma_kernel<<<96, 32, 0, stream>>>(x, W_write, b_write, W_wq, b_wq,
                                            W_rq, b_rq, mv, wq, rq);
    // 2) write logits (streams 128 MB from HBM, fills L2)
    wlogits_kernel<<<BATCH * 16, 256, 0, stream>>>(memory, wq, wl);
    // 3) write softmax stats
    softmax_stats_kernel<<<BATCH, 256, 0, stream>>>(wl, wstat);
    // 4) read logits over on-the-fly new_mem (L2-resident pass)
    rlogits_kernel<<<BATCH * 16, 256, 0, stream>>>(memory, mv, rq, wl, wstat, rl);
    // 5) read softmax stats
    softmax_stats_kernel<<<BATCH, 256, 0, stream>>>(rl, rstat);
    // 6) weighted readout (L2-resident pass)
    readout_kernel<<<BATCH, 256, 0, stream>>>(memory, mv, wl, wstat, rl, rstat, ro);
    // 7) output projection via WMMA
    out_wmma_kernel<<<64, 32, 0, stream>>>(ro, W_ro, b_ro, out);
}